// NeRFRenderer_31954556682972
// MI455X (gfx1250) — compile-verified
//
#include <hip/hip_runtime.h>
#include <stdint.h>

// ---------------------------------------------------------------------------
// NeRF fine-resampling + volume rendering for MI455X (gfx1250, wave32).
// One ray per wave: 128 coarse weights = 4 f32/lane, 64 fine samples = 2/lane.
// Coarse-weight row staged to LDS with gfx1250 async-to-LDS (b128/lane,
// TH_LOAD_NT), overlapped with AABB intersection math; CDF built via wave32
// shuffle scan. Memory-bound problem (~470MB traffic -> ~20us @ 23.3TB/s);
// single-use streams are marked non-temporal to avoid churning the 192MB L2.
// WMMA is not applicable (no dense matmul; VALU work is ~0.13G ops total).
// ---------------------------------------------------------------------------

#define N_RAYS 262144
#define T0 128          // coarse steps
#define TF 64           // fine steps
#define RAYS_PER_BLOCK 8
#define THREADS (RAYS_PER_BLOCK * 32)
#define OUT_COLS (4 * TF + 3)   // 259

__device__ __forceinline__ float wave_incl_scan(float x) {
  const int lane = threadIdx.x & 31;
  #pragma unroll
  for (int d = 1; d < 32; d <<= 1) {
    float y = __shfl_up(x, d, 32);
    if (lane >= d) x += y;
  }
  return x;
}

__device__ __forceinline__ float wave_sum(float x) {
  #pragma unroll
  for (int d = 16; d > 0; d >>= 1) x += __shfl_xor(x, d, 32);
  return x;
}

__global__ __launch_bounds__(THREADS) void nerf_main_kernel(
    const float* __restrict__ rays_o, const float* __restrict__ rays_d,
    const float* __restrict__ aabb,   const float* __restrict__ wc,
    const float* __restrict__ sig,    float* __restrict__ out,
    float* __restrict__ loss_part)
{
  __shared__ __align__(16) float s_w[RAYS_PER_BLOCK][T0];      // coarse weights (async landed)
  __shared__ float s_cdf[RAYS_PER_BLOCK][T0 + 1];              // normalized clamped cdf
  __shared__ float s_cw [RAYS_PER_BLOCK][T0 + 1];              // raw cumsum (for interlevel loss)
  __shared__ float s_bf [RAYS_PER_BLOCK][TF + 1];              // fine bins (warped space)
  __shared__ float s_rb [RAYS_PER_BLOCK][TF + 1];              // real-space bins
  __shared__ float s_red[RAYS_PER_BLOCK];

  const int lane = threadIdx.x & 31;
  const int wv   = threadIdx.x >> 5;
  const int ray  = blockIdx.x * RAYS_PER_BLOCK + wv;

  // ---- async stage: coarse-weight row -> LDS (16B per lane, 512B per wave).
  //      Single-use stream -> non-temporal load policy.
  {
    unsigned lds_addr = (unsigned)(uintptr_t)&s_w[wv][lane * 4];
    unsigned long long gaddr =
        (unsigned long long)(uintptr_t)(wc + (size_t)ray * T0 + lane * 4);
    asm volatile("global_load_async_to_lds_b128 %0, %1, off th:TH_LOAD_NT"
                 :: "v"(lds_addr), "v"(gaddr) : "memory");
  }
  // prefetch sigma row (needed later) while we do ray/AABB math
  __builtin_prefetch(sig + (size_t)ray * TF + lane, 0, 0);

  // ---- near/far from AABB (all lanes redundantly; broadcast loads) ----
  const float ox = rays_o[ray * 3 + 0], oy = rays_o[ray * 3 + 1], oz = rays_o[ray * 3 + 2];
  const float dx = rays_d[ray * 3 + 0], dy = rays_d[ray * 3 + 1], dz = rays_d[ray * 3 + 2];
  const float inx = 1.0f / (dx + 1e-15f);
  const float iny = 1.0f / (dy + 1e-15f);
  const float inz = 1.0f / (dz + 1e-15f);
  float t0x = (aabb[0] - ox) * inx, t1x = (aabb[3] - ox) * inx;
  float t0y = (aabb[1] - oy) * iny, t1y = (aabb[4] - oy) * iny;
  float t0z = (aabb[2] - oz) * inz, t1z = (aabb[5] - oz) * inz;
  float mnx = (t0x < t1x) ? t0x : t1x, mxx = (t0x > t1x) ? t0x : t1x;
  float mny = (t0y < t1y) ? t0y : t1y, mxy = (t0y > t1y) ? t0y : t1y;
  float mnz = (t0z < t1z) ? t0z : t1z, mxz = (t0z > t1z) ? t0z : t1z;
  float nearv = fmaxf(fmaxf(mnx, mny), mnz);
  float farv  = fminf(fminf(mxx, mxy), mxz);
  if (farv < nearv) { nearv = 1e9f; farv = 1e9f; }
  nearv = fmaxf(nearv, 0.05f);
  const float s_near = (nearv < 1.0f) ? nearv * 0.5f : 1.0f - 1.0f / (2.0f * nearv);
  const float s_far  = (farv  < 1.0f) ? farv  * 0.5f : 1.0f - 1.0f / (2.0f * farv);

  // ---- wait for async landing, then build CDF + raw cumsum in LDS ----
  asm volatile("s_wait_asynccnt 0" ::: "memory");
  __syncthreads();

  float w0 = s_w[wv][lane * 4 + 0];
  float w1 = s_w[wv][lane * 4 + 1];
  float w2 = s_w[wv][lane * 4 + 2];
  float w3 = s_w[wv][lane * 4 + 3];
  // adjusted (w + 0.01) in-lane inclusive sums
  float c1 = w0 + 0.01f, c2 = c1 + (w1 + 0.01f), c3 = c2 + (w2 + 0.01f), c4 = c3 + (w3 + 0.01f);
  // raw in-lane inclusive sums
  float r1 = w0, r2 = r1 + w1, r3 = r2 + w2, r4 = r3 + w3;
  float inclA = wave_incl_scan(c4);
  float inclR = wave_incl_scan(r4);
  float exclA = inclA - c4;
  float exclR = inclR - r4;
  float total = __shfl(inclA, 31, 32);
  float invT  = 1.0f / total;
  s_cdf[wv][lane * 4 + 1] = fminf((exclA + c1) * invT, 1.0f);
  s_cdf[wv][lane * 4 + 2] = fminf((exclA + c2) * invT, 1.0f);
  s_cdf[wv][lane * 4 + 3] = fminf((exclA + c3) * invT, 1.0f);
  s_cdf[wv][lane * 4 + 4] = fminf((exclA + c4) * invT, 1.0f);
  s_cw [wv][lane * 4 + 1] = exclR + r1;
  s_cw [wv][lane * 4 + 2] = exclR + r2;
  s_cw [wv][lane * 4 + 3] = exclR + r3;
  s_cw [wv][lane * 4 + 4] = exclR + r4;
  if (lane == 0) { s_cdf[wv][0] = 0.0f; s_cw[wv][0] = 0.0f; }
  __syncthreads();

  // ---- inverse-CDF sample 65 fine bins; map to real space ----
  for (int j = lane; j <= TF; j += 32) {
    const float u = (j + 0.5f) * (1.0f / (float)(TF + 1));
    int lo = 0, hi = T0 + 1;                    // upper_bound over cdf[0..128]
    while (lo < hi) {
      int mid = (lo + hi) >> 1;
      if (s_cdf[wv][mid] <= u) lo = mid + 1; else hi = mid;
    }
    int below = lo - 1; below = (below < 0) ? 0 : ((below > T0) ? T0 : below);
    int above = (lo > T0) ? T0 : lo;
    const float cg0 = s_cdf[wv][below];
    const float cg1 = s_cdf[wv][above];
    const float b0  = (float)below * (1.0f / (float)T0);   // bins0[k] = k/128 exact
    const float b1  = (float)above * (1.0f / (float)T0);
    float t = (u - cg0) / (cg1 - cg0);
    if (!(t == t)) t = 0.0f;                    // nan_to_num
    t = fminf(fmaxf(t, 0.0f), 1.0f);
    const float bf = b0 + t * (b1 - b0);
    s_bf[wv][j] = bf;
    const float xw = s_near * (1.0f - bf) + s_far * bf;     // warped
    s_rb[wv][j] = (xw < 0.5f) ? 2.0f * xw : 1.0f / (2.0f - 2.0f * xw);
  }
  __syncthreads();

  // ---- per-sample: xyz + contraction, ds, transmittance scan, weights ----
  float* orow = out + (size_t)ray * OUT_COLS;
  float ds_scan[2], ds_alpha[2], tmid[2], wgt[2];

  #pragma unroll
  for (int k = 0; k < 2; k++) {
    const int j = lane + 32 * k;
    const float rb0 = s_rb[wv][j], rb1 = s_rb[wv][j + 1];
    const float t_mid = 0.5f * (rb0 + rb1);
    const float delta = rb1 - rb0;
    const float sgm = __builtin_nontemporal_load(sig + (size_t)ray * TF + j);
    float d_s = delta * sgm;
    ds_scan[k]  = (j == TF - 1) ? 0.0f : d_s;             // last excluded from cumsum
    ds_alpha[k] = (j == TF - 1) ? __builtin_inff() : d_s; // 'last_sample' background
    tmid[k] = t_mid;

    // contraction (MERF): scale argmax-|coord| axis specially when |x|inf >= 1
    const float X = ox + dx * t_mid, Y = oy + dy * t_mid, Z = oz + dz * t_mid;
    const float aX = fabsf(X), aY = fabsf(Y), aZ = fabsf(Z);
    const float mag = fmaxf(fmaxf(aX, aY), aZ);
    float cX = X, cY = Y, cZ = Z;
    if (mag >= 1.0f) {
      const float im = 1.0f / mag;
      const float sm = (2.0f - im) * im;
      if (aX == mag)      { cX = X * sm; cY = Y * im; cZ = Z * im; }
      else if (aY == mag) { cX = X * im; cY = Y * sm; cZ = Z * im; }
      else                { cX = X * im; cY = Y * im; cZ = Z * sm; }
    }
    __builtin_nontemporal_store(cX, orow + 3 * j + 0);
    __builtin_nontemporal_store(cY, orow + 3 * j + 1);
    __builtin_nontemporal_store(cZ, orow + 3 * j + 2);
  }

  // exclusive scan of ds over 64 samples (lane holds j and j+32)
  float inc0 = wave_incl_scan(ds_scan[0]);
  float tot0 = __shfl(inc0, 31, 32);
  float inc1 = wave_incl_scan(ds_scan[1]);
  float excl[2];
  excl[0] = inc0 - ds_scan[0];
  excl[1] = tot0 + inc1 - ds_scan[1];

  float wsum_l = 0.0f, depth_l = 0.0f;
  #pragma unroll
  for (int k = 0; k < 2; k++) {
    const int j = lane + 32 * k;
    const float alpha = 1.0f - expf(-ds_alpha[k]);
    const float trans = expf(-excl[k]);
    float w = alpha * trans;
    if (!(w == w)) w = 0.0f;                    // nan_to_num
    wgt[k] = w;
    __builtin_nontemporal_store(w, orow + 3 * TF + j);   // weights cols 192..255
    wsum_l += w;
    depth_l += w * tmid[k];
  }
  const float wsum  = wave_sum(wsum_l);
  const float depth = wave_sum(depth_l);
  if (lane == 0) {
    __builtin_nontemporal_store(depth, orow + 4 * TF + 0);  // col 256
    __builtin_nontemporal_store(wsum,  orow + 4 * TF + 1);  // col 257
  }

  // ---- interlevel (proposal) loss: searchsorted on uniform grid is floor() ----
  float lloss = 0.0f;
  #pragma unroll
  for (int k = 0; k < 2; k++) {
    const int j = lane + 32 * k;
    const float v_lo = s_bf[wv][j];
    const float v_hi = s_bf[wv][j + 1];
    int ilo = (int)floorf(v_lo * (float)T0);
    ilo = (ilo < 0) ? 0 : ((ilo > T0 - 1) ? T0 - 1 : ilo);
    int ihi = (int)floorf(v_hi * (float)T0);
    ihi = (ihi < 0) ? 0 : ((ihi > T0 - 1) ? T0 - 1 : ihi);
    const float w = s_cw[wv][ihi + 1] - s_cw[wv][ilo];
    float d0 = wgt[k] - w;
    d0 = fmaxf(d0, 0.0f);
    lloss += d0 * d0 / (wgt[k] + 1e-8f);
  }
  const float lsum = wave_sum(lloss);
  if (lane == 0) s_red[wv] = lsum;
  __syncthreads();
  if (threadIdx.x == 0) {
    float t = 0.0f;
    #pragma unroll
    for (int i = 0; i < RAYS_PER_BLOCK; i++) t += s_red[i];
    loss_part[blockIdx.x] = t;   // re-read by reduce kernel: keep RT policy
  }
}

__global__ __launch_bounds__(256) void reduce_loss_kernel(
    const float* __restrict__ part, float* __restrict__ scalar, int nparts)
{
  __shared__ float red[8];
  float s = 0.0f;
  for (int i = threadIdx.x; i < nparts; i += 256) s += part[i];
  s = wave_sum(s);
  const int lane = threadIdx.x & 31, wv = threadIdx.x >> 5;
  if (lane == 0) red[wv] = s;
  __syncthreads();
  if (threadIdx.x == 0) {
    float t = 0.0f;
    #pragma unroll
    for (int i = 0; i < 8; i++) t += red[i];
    *scalar = t * (1.0f / ((float)N_RAYS * (float)TF));
  }
}

__global__ __launch_bounds__(256) void fill_loss_kernel(
    float* __restrict__ out, const float* __restrict__ scalar)
{
  const int n = blockIdx.x * 256 + threadIdx.x;
  if (n < N_RAYS)
    __builtin_nontemporal_store(*scalar, out + (size_t)n * OUT_COLS + (4 * TF + 2));
}

extern "C" void kernel_launch(void* const* d_in, const int* in_sizes, int n_in,
                              void* d_out, int out_size, void* d_ws, size_t ws_size,
                              hipStream_t stream) {
  const float* rays_o = (const float*)d_in[0];
  const float* rays_d = (const float*)d_in[1];
  const float* aabb   = (const float*)d_in[2];
  const float* wc     = (const float*)d_in[3];
  const float* sig    = (const float*)d_in[4];
  float* out = (float*)d_out;

  const int nblocks = N_RAYS / RAYS_PER_BLOCK;      // 32768
  float* part   = (float*)d_ws;                     // [nblocks]
  float* scalar = part + nblocks;                   // [1]

  nerf_main_kernel<<<dim3(nblocks), dim3(THREADS), 0, stream>>>(
      rays_o, rays_d, aabb, wc, sig, out, part);
  reduce_loss_kernel<<<dim3(1), dim3(256), 0, stream>>>(part, scalar, nblocks);
  fill_loss_kernel<<<dim3(N_RAYS / 256), dim3(256), 0, stream>>>(out, scalar);
}